// HardMoEClassifier_17463337026021
// MI455X (gfx1250) — compile-verified
//
#include <hip/hip_runtime.h>

typedef float v2f __attribute__((ext_vector_type(2)));
typedef float v8f __attribute__((ext_vector_type(8)));

#define Hdim 1024
#define Edim 6
#define Ldim 2
#define NW 18              // 6 gate rows + 12 expert rows
#define WROWS 19           // + one zero row for N-padding (cols 18..31)
#define WSTRIDE 1026       // floats; stride%64==2 -> conflict-free b64 reads, 8B aligned
#define OSTRIDE 33         // padded output tile stride
#define WAVES 8
#define ROWS_PER_BLOCK 128 // 8 waves * 16 rows

__global__ __launch_bounds__(256) void hardmoe_wmma_kernel(
    const float* __restrict__ cls,     // [B, H]
    const float* __restrict__ gate_w,  // [E, H]
    const float* __restrict__ gate_b,  // [E]
    const float* __restrict__ exp_w,   // [E, L, H] -> [12, H]
    const float* __restrict__ exp_b,   // [E, L]
    float* __restrict__ out)           // [B, L] fp32
{
    extern __shared__ float smem[];
    float* lds_w = smem;                         // WROWS * WSTRIDE
    float* lds_o = smem + WROWS * WSTRIDE;       // WAVES * 16 * OSTRIDE

    const int tid = threadIdx.x;

    // ---- Stage combined weights Wc[18][1024] into LDS (+ zero row 18) ----
    for (int i = tid; i < NW * Hdim; i += 256) {
        const int r = i >> 10;
        const int c = i & (Hdim - 1);
        const float v = (r < Edim) ? gate_w[r * Hdim + c]
                                   : exp_w[(r - Edim) * Hdim + c];
        lds_w[r * WSTRIDE + c] = v;
    }
    for (int c = tid; c < Hdim; c += 256) lds_w[NW * WSTRIDE + c] = 0.0f;
    __syncthreads();

    const int wave = tid >> 5;
    const int lane = tid & 31;
    const int half = lane >> 4;     // selects K pair {0,1} vs {2,3}
    const int l16  = lane & 15;

    const int m0 = blockIdx.x * ROWS_PER_BLOCK + wave * 16;

    // A fragment source: row M = l16, columns k + 2*half + {0,1}
    const float* arow = cls + ((size_t)(m0 + l16) << 10) + (half << 1);
    // B fragments: tile0 col n = l16, tile1 col n = 16+l16 (clamped to zero row)
    const float* b0p = &lds_w[l16 * WSTRIDE + (half << 1)];
    int n1 = 16 + l16; if (n1 > NW) n1 = NW;
    const float* b1p = &lds_w[n1 * WSTRIDE + (half << 1)];

    v8f c0 = {};
    v8f c1 = {};

    #pragma unroll 4
    for (int k = 0; k < Hdim; k += 4) {
        const v2f a  = *(const v2f*)(arow + k);
        const v2f b0 = *(const v2f*)(b0p + k);
        const v2f b1 = *(const v2f*)(b1p + k);
        c0 = __builtin_amdgcn_wmma_f32_16x16x4_f32(false, a, false, b0,
                                                   (short)0, c0, false, false);
        c1 = __builtin_amdgcn_wmma_f32_16x16x4_f32(false, a, false, b1,
                                                   (short)0, c1, false, false);
    }

    // ---- Scatter C tiles to LDS: VGPR r -> (M = r + 8*half, N = l16 [+16]) ----
    float* ow = &lds_o[wave * 16 * OSTRIDE];
    #pragma unroll
    for (int r = 0; r < 8; ++r) {
        const int m = r + (half << 3);
        ow[m * OSTRIDE + l16]      = c0[r];
        ow[m * OSTRIDE + 16 + l16] = c1[r];
    }
    __syncthreads();

    // ---- Gate argmax + expert select + bias, 16 lanes per wave ----
    if (lane < 16) {
        const float* row = &ow[lane * OSTRIDE];
        int   best  = 0;
        float bestv = row[0] + gate_b[0];
        #pragma unroll
        for (int e = 1; e < Edim; ++e) {
            const float v = row[e] + gate_b[e];
            if (v > bestv) { bestv = v; best = e; }   // strict > == argmax first-max
        }
        const float o0 = row[Edim + best * 2 + 0] + exp_b[best * 2 + 0];
        const float o1 = row[Edim + best * 2 + 1] + exp_b[best * 2 + 1];
        const size_t gi = (size_t)(m0 + lane) * Ldim;
        out[gi + 0] = o0;
        out[gi + 1] = o1;
    }
}

extern "C" void kernel_launch(void* const* d_in, const int* in_sizes, int n_in,
                              void* d_out, int out_size, void* d_ws, size_t ws_size,
                              hipStream_t stream) {
    const float* cls = (const float*)d_in[0];   // [B, H]
    const float* gw  = (const float*)d_in[1];   // [E, H]
    const float* gb  = (const float*)d_in[2];   // [E]
    const float* ew  = (const float*)d_in[3];   // [E, L, H]
    const float* eb  = (const float*)d_in[4];   // [E, L]
    float* out = (float*)d_out;

    const int B = in_sizes[0] / Hdim;           // 65536
    const int nblk = B / ROWS_PER_BLOCK;        // 512

    const size_t lds_bytes =
        (size_t)(WROWS * WSTRIDE + WAVES * 16 * OSTRIDE) * sizeof(float);

    hipLaunchKernelGGL(hardmoe_wmma_kernel, dim3(nblk), dim3(256), lds_bytes,
                       stream, cls, gw, gb, ew, eb, out);
}